// RNNTLoss_88141318848782
// MI455X (gfx1250) — compile-verified
//
#include <hip/hip_runtime.h>
#include <hip/hip_bf16.h>
#include <math.h>

// Problem shape (fixed by the reference)
#define B_  8
#define T_  256
#define U_  128
#define U1_ 129
#define V_  256

typedef float        v4f __attribute__((ext_vector_type(4)));
typedef unsigned int v4u __attribute__((ext_vector_type(4)));
typedef int          v8i __attribute__((ext_vector_type(8)));
typedef int          v4i __attribute__((ext_vector_type(4)));

// ---------------------------------------------------------------------------
// Phase 1: log-softmax reduction. One wave32 per (b,t,u) row of V=256 floats.
// Bandwidth-bound: 270 MB streamed once with non-temporal b128 loads.
// Emits blank_lp (B,T,129) and lab_lp (B,T,128) into workspace.
// ---------------------------------------------------------------------------
__global__ __launch_bounds__(256) void rnnt_prep(
    const float* __restrict__ logits,
    const int*   __restrict__ targets,
    float*       __restrict__ blank_lp,
    float*       __restrict__ lab_lp)
{
  const int gtid = blockIdx.x * blockDim.x + threadIdx.x;
  const int row  = gtid >> 5;               // one wave per row
  const int lane = threadIdx.x & 31;
  if (row >= B_ * T_ * U1_) return;

  const float* rp = logits + (size_t)row * V_;
  // lane covers v = lane*4..lane*4+3 and v = 128+lane*4..+3
  v4f x0 = __builtin_nontemporal_load((const v4f*)rp + lane);
  v4f x1 = __builtin_nontemporal_load((const v4f*)rp + 32 + lane);

  float m = fmaxf(fmaxf(fmaxf(x0.x, x0.y), fmaxf(x0.z, x0.w)),
                  fmaxf(fmaxf(x1.x, x1.y), fmaxf(x1.z, x1.w)));
  for (int o = 16; o > 0; o >>= 1) m = fmaxf(m, __shfl_xor(m, o, 32));

  float s = expf(x0.x - m) + expf(x0.y - m) + expf(x0.z - m) + expf(x0.w - m)
          + expf(x1.x - m) + expf(x1.y - m) + expf(x1.z - m) + expf(x1.w - m);
  for (int o = 16; o > 0; o >>= 1) s += __shfl_xor(s, o, 32);

  const float logZ = m + logf(s);

  const int u  = row % U1_;
  const int bt = row / U1_;

  if (lane == 0) blank_lp[row] = x0.x - logZ;   // lane 0's x0.x is v==0 (BLANK)

  if (u < U_) {
    const int b    = row / (T_ * U1_);
    const int tgt  = targets[b * U_ + u];       // wave-uniform
    const int comp = tgt & 3;
    float v0 = (comp == 0) ? x0.x : (comp == 1) ? x0.y : (comp == 2) ? x0.z : x0.w;
    float v1 = (comp == 0) ? x1.x : (comp == 1) ? x1.y : (comp == 2) ? x1.z : x1.w;
    float g0 = __shfl(v0, (tgt >> 2) & 31, 32);
    float g1 = __shfl(v1, ((tgt - 128) >> 2) & 31, 32);
    float xv = (tgt < 128) ? g0 : g1;
    if (lane == 0) lab_lp[bt * U_ + u] = xv - logZ;
  }
}

// ---------------------------------------------------------------------------
// TDM: 1-D tile DMA global -> LDS via Tensor Data Mover (D# per ISA 8.3/8.4).
// group0: count=1, lds_addr, 57-bit global addr, type=2.
// group1: data_size=4B (bits 17:16 = 2), tensor_dim0 = tile_dim0 = nelem,
//         tensor_dim1 = tile_dim1 = 1, dim0_stride = nelem.
// Toolchain uses the 6-arg builtin: (v4u, v8i, v4i, v4i, v8i, i32 cpol);
// groups 2/3 and the trailing group are unused for <=2-D tensors -> zeros.
// ---------------------------------------------------------------------------
__device__ __forceinline__ void tdm_load_1d(const void* gptr,
                                            unsigned lds_byte_off,
                                            unsigned nelem)
{
  unsigned long long ga = (unsigned long long)gptr;
  v4u g0;
  g0.x = 1u;                                              // count=1, user mode
  g0.y = lds_byte_off;                                    // lds_addr (bytes)
  g0.z = (unsigned)(ga & 0xffffffffu);                    // global_addr[31:0]
  g0.w = (unsigned)((ga >> 32) & 0x1ffffffu) | (2u << 30);// addr[56:32] | type=2
  v8i g1;
  g1[0] = (int)(2u << 16);                                // data_size = 4 bytes
  g1[1] = (int)((nelem & 0xffffu) << 16);                 // tensor_dim0[15:0]
  g1[2] = (int)(((nelem >> 16) & 0xffffu) | (1u << 16));  // dim0[31:16] | tensor_dim1=1
  g1[3] = (int)((nelem & 0xffffu) << 16);                 // tile_dim0
  g1[4] = 1;                                              // tile_dim1=1, tile_dim2=0
  g1[5] = (int)nelem;                                     // dim0_stride lo
  g1[6] = 0;                                              // dim0_stride hi | dim1_stride lo
  g1[7] = 0;
  v4i z4 = {0, 0, 0, 0};
  v8i z8 = {0, 0, 0, 0, 0, 0, 0, 0};
  __builtin_amdgcn_tensor_load_to_lds(g0, g1, z4, z4, z8, 0);
}

// ---------------------------------------------------------------------------
// Phase 2: anti-diagonal DP, one workgroup per batch element.
// blank (129*256 f32) + lab (128*256 f32) = 263 KB -> fits 320 KB WGP LDS,
// DMA'd in with two TDM descriptors. 384 diagonals, 1 barrier each.
//   alpha[t,u] = logaddexp(alpha[t-1,u] + blank[t-1,u],
//                          alpha[t,u-1] + lab[t,u-1])
// Diagonal d holds A_d[u] = alpha[d-u, u]; both parents live on diagonal d-1.
// ---------------------------------------------------------------------------
#define LDS_BLANK_ELEMS (T_ * U1_)          // 33024
#define LDS_LAB_ELEMS   (T_ * U_)           // 32768
#define LDS_BUF_OFF     (LDS_BLANK_ELEMS + LDS_LAB_ELEMS)

__global__ __launch_bounds__(160) void rnnt_dp(
    const float* __restrict__ blank_lp,
    const float* __restrict__ lab_lp,
    const int*   __restrict__ input_lengths,
    const int*   __restrict__ target_lengths,
    float*       __restrict__ loss_out)
{
  extern __shared__ float smem[];
  float* blankS = smem;                       // [T_][U1_]
  float* labS   = smem + LDS_BLANK_ELEMS;     // [T_][U_]
  float* buf0   = smem + LDS_BUF_OFF;         // diagonal buffers (132 each)
  float* buf1   = buf0 + 132;

  const int b   = blockIdx.x;
  const int tid = threadIdx.x;

  if (tid < 32) {   // wave 0 issues both async tensor DMAs and waits
    tdm_load_1d(blank_lp + (size_t)b * LDS_BLANK_ELEMS, 0u,
                (unsigned)LDS_BLANK_ELEMS);
    tdm_load_1d(lab_lp + (size_t)b * LDS_LAB_ELEMS,
                (unsigned)(LDS_BLANK_ELEMS * sizeof(float)),
                (unsigned)LDS_LAB_ELEMS);
    __builtin_amdgcn_s_wait_tensorcnt(0);
  }
  if (tid == 0) buf0[0] = 0.0f;               // alpha[0,0] = 0
  __syncthreads();

  const int t_last = input_lengths[b] - 1;    // in [127,255]
  const int u_last = target_lengths[b];       // in [64,128]
  const int d_cap  = t_last + u_last;

  float* prev = buf0;
  float* cur  = buf1;
  const int u = tid;

  for (int d = 1; d <= (T_ - 1) + U_; ++d) {
    if (u <= U_ && u <= d && (d - u) <= (T_ - 1)) {
      const int t = d - u;
      float up = (t >= 1) ? prev[u] + blankS[(t - 1) * U1_ + u] : -INFINITY;
      float lf = (u >= 1) ? prev[u - 1] + labS[t * U_ + (u - 1)] : -INFINITY;
      float m  = fmaxf(up, lf);
      float val = (m > -INFINITY)
                    ? m + logf(expf(up - m) + expf(lf - m))
                    : -INFINITY;
      cur[u] = val;
      if (d == d_cap && u == u_last) {
        loss_out[b] = -(val + blankS[t_last * U1_ + u_last]);
      }
    }
    __syncthreads();
    float* tmp = prev; prev = cur; cur = tmp;
  }
}

// ---------------------------------------------------------------------------
// Phase 3: deterministic mean over batch (fixed summation order).
// ---------------------------------------------------------------------------
__global__ void rnnt_finalize(const float* __restrict__ loss_arr,
                              float* __restrict__ out)
{
  if (blockIdx.x == 0 && threadIdx.x == 0) {
    float s = 0.0f;
    for (int i = 0; i < B_; ++i) s += loss_arr[i];
    out[0] = s / (float)B_;
  }
}

// ---------------------------------------------------------------------------
extern "C" void kernel_launch(void* const* d_in, const int* in_sizes, int n_in,
                              void* d_out, int out_size, void* d_ws, size_t ws_size,
                              hipStream_t stream)
{
  const float* logits         = (const float*)d_in[0];
  const int*   targets        = (const int*)d_in[1];
  const int*   input_lengths  = (const int*)d_in[2];
  const int*   target_lengths = (const int*)d_in[3];

  float* blank_lp = (float*)d_ws;                       // B*T*129 floats
  float* lab_lp   = blank_lp + (size_t)B_ * T_ * U1_;   // B*T*128 floats
  float* loss_arr = lab_lp + (size_t)B_ * T_ * U_;      // B floats

  // Phase 1: one wave per row; 8 waves / 256-thread block.
  const int rows   = B_ * T_ * U1_;                     // 264192
  const int blocks = rows / 8;                          // 33024
  rnnt_prep<<<blocks, 256, 0, stream>>>(logits, targets, blank_lp, lab_lp);

  // Phase 2: one workgroup (5 waves) per batch element; 263 KB dynamic LDS.
  const size_t shmem = (size_t)(LDS_BUF_OFF + 2 * 132) * sizeof(float);
  rnnt_dp<<<B_, 160, shmem, stream>>>(blank_lp, lab_lp,
                                      input_lengths, target_lengths, loss_arr);

  rnnt_finalize<<<1, 32, 0, stream>>>(loss_arr, (float*)d_out);
}